// HarmonicExcitationGenerator_50105088475734
// MI455X (gfx1250) — compile-verified
//
#include <hip/hip_runtime.h>
#include <string.h>

// ---------------------------------------------------------------------------
// HarmonicExcitationGenerator for MI455X (gfx1250, wave32)
//   B=4, F=128, BLOCK=2048, N_HARM=64, SR=16000
//   out = [signal(B*T), noise(B*T), ln(B*T)]  (T = F*BLOCK = 262144)
// ---------------------------------------------------------------------------

#define BATCH      4
#define FRAMES     128
#define BLOCK_SZ   2048
#define NHARM      64
#define T_PER_B    (FRAMES * BLOCK_SZ)      // 262144
#define N_TOTAL    (BATCH * T_PER_B)        // 1048576
#define TWO_PI_F   6.2831853071795864769f
#define INV_2PI_F  0.15915494309189533577f
#define GLOBAL_AMP 0.35f

typedef float v2f __attribute__((ext_vector_type(2)));
typedef float v8f __attribute__((ext_vector_type(8)));

typedef __attribute__((address_space(1))) int* gint_p;   // global (device) ptr
typedef __attribute__((address_space(3))) int* lint_p;   // LDS ptr

// workspace layout (floats): dphi[512] | base[512] | ap[512] | loud[512] | ln[512] | ha[512*64]
#define WS_DPHI 0
#define WS_BASE 512
#define WS_AP   1024
#define WS_LOUD 1536
#define WS_LN   2048
#define WS_HA   2560

// ---------------------------------------------------------------------------
// Kernel 1: per-frame prep + phase-base scan.  grid = B, block = F threads.
// ---------------------------------------------------------------------------
__global__ __launch_bounds__(FRAMES) void heg_prep_kernel(
    const float* __restrict__ pitch, const float* __restrict__ harm,
    const float* __restrict__ per,   const float* __restrict__ loud,
    const float* __restrict__ lnorm, float* __restrict__ ws) {
  const int b  = blockIdx.x;
  const int f  = threadIdx.x;
  const int bf = b * FRAMES + f;

  float pr = per[bf];
  float uv = (pr > 0.5f) ? 1.0f : 0.0f;
  float p  = pitch[bf];
  p = fmaxf(p, 0.001f);
  if (isnan(p) || isinf(p)) p = 0.0f;       // nan_to_num(nan/±inf -> 0)
  p *= uv;

  const float dphi = TWO_PI_F * p * (1.0f / 16000.0f);

  // exclusive scan of per-frame phase increments (Hillis-Steele in LDS)
  __shared__ float sb[FRAMES];
  const float v = dphi * (float)BLOCK_SZ;
  sb[f] = v;
  __syncthreads();
  for (int off = 1; off < FRAMES; off <<= 1) {
    float o = (f >= off) ? sb[f - off] : 0.0f;
    __syncthreads();
    sb[f] += o;
    __syncthreads();
  }
  const float base = sb[f] - v;             // exclusive prefix

  // harmonic amp anti-alias + normalize (two passes, no big register array)
  float sum = 0.0f;
  for (int k = 0; k < NHARM; ++k) {
    float a  = harm[bf * NHARM + k];
    float aa = ((p * (float)(k + 1)) < 8000.0f) ? 1.0f : 0.0f;
    sum += a * (aa + 1e-4f);
  }
  const float inv = 1.0f / sum;
  for (int k = 0; k < NHARM; ++k) {
    float a  = harm[bf * NHARM + k];
    float aa = ((p * (float)(k + 1)) < 8000.0f) ? 1.0f : 0.0f;
    ws[WS_HA + bf * NHARM + k] = a * (aa + 1e-4f) * inv;
  }

  ws[WS_DPHI + bf] = dphi;
  ws[WS_BASE + bf] = base;
  ws[WS_AP   + bf] = (1.0f - pr) * uv;
  ws[WS_LOUD + bf] = loud[bf];
  ws[WS_LN   + bf] = lnorm[bf];
}

// ---------------------------------------------------------------------------
// threefry2x32-20 (JAX-compatible), key = jax.random.key(42) -> (0, 42)
// ---------------------------------------------------------------------------
__device__ __forceinline__ unsigned heg_threefry_bits(unsigned g) {
  const unsigned N2 = (unsigned)(N_TOTAL / 2);     // 524288
  const unsigned i  = (g < N2) ? g : (g - N2);
  unsigned X0 = i;                                 // counts[i]
  unsigned X1 = N2 + i;                            // counts[N/2 + i]
  const unsigned ks0 = 0u, ks1 = 42u, ks2 = 0u ^ 42u ^ 0x1BD11BDAu;
  X0 += ks0; X1 += ks1;
#define HEG_RND(r) { X0 += X1; X1 = (X1 << (r)) | (X1 >> (32 - (r))); X1 ^= X0; }
  HEG_RND(13) HEG_RND(15) HEG_RND(26) HEG_RND(6)   X0 += ks1; X1 += ks2 + 1u;
  HEG_RND(17) HEG_RND(29) HEG_RND(16) HEG_RND(24)  X0 += ks2; X1 += ks0 + 2u;
  HEG_RND(13) HEG_RND(15) HEG_RND(26) HEG_RND(6)   X0 += ks0; X1 += ks1 + 3u;
  HEG_RND(17) HEG_RND(29) HEG_RND(16) HEG_RND(24)  X0 += ks1; X1 += ks2 + 4u;
  HEG_RND(13) HEG_RND(15) HEG_RND(26) HEG_RND(6)   X0 += ks2; X1 += ks0 + 5u;
#undef HEG_RND
  return (g < N2) ? X0 : X1;
}

// ---------------------------------------------------------------------------
// One 16-sample tile: stride-4 Chebyshev recurrence (VALU) feeding a chain of
// 16 x V_WMMA_F32_16X16X4_F32 (co-executing on the matrix pipe), followed by
// the C/D-layout extraction.  Returns, in lane L, the harmonic sum for
// M = (L>=16 ? 8 : 0) + (L & 7)  (pre-shuffle value).
// ---------------------------------------------------------------------------
__device__ __forceinline__ float heg_tile_sinsum(float base, float dphi, int s,
                                                 bool hi, int lane,
                                                 const v2f (&bq)[16]) {
  float omega = base + (float)(s + 1) * dphi;
  omega -= TWO_PI_F * floorf(omega * INV_2PI_F);    // exact for all harmonics

  const float s1 = __sinf(omega);
  const float c1 = __cosf(omega);
  const float s2 = 2.0f * c1 * s1;
  const float s3 = 2.0f * c1 * s2 - s1;
  const float s4 = 2.0f * c1 * s3 - s2;
  const float c2 = 2.0f * c1 * c1 - 1.0f;
  const float c4 = 2.0f * c2 * c2 - 1.0f;
  const float stp = 2.0f * c4;                      // stride-4 recurrence coeff

  // A 16x4 f32: lanes 0-15 -> (K0,K1) = (s_{4c+1}, s_{4c+2});
  //             lanes 16-31 -> (K2,K3) = (s_{4c+3}, s_{4c+4})
  float cur0 = hi ?  s3 : s1;
  float cur1 = hi ?  s4 : s2;
  float prv0 = hi ? -s1 : -s3;                      // s_{k-4} seeds (sin(-x) = -sin x)
  float prv1 = hi ? 0.0f : -s2;

  v8f acc = {0.f, 0.f, 0.f, 0.f, 0.f, 0.f, 0.f, 0.f};
#pragma unroll
  for (int c = 0; c < 16; ++c) {
    v2f a; a.x = cur0; a.y = cur1;
    acc = __builtin_amdgcn_wmma_f32_16x16x4_f32(
        /*neg_a=*/false, a, /*neg_b=*/false, bq[c],
        /*c_mod=*/(short)0, acc, /*reuse_a=*/false, /*reuse_b=*/false);
    // advance both K-slots by 4 harmonics: s_{k+4} = 2cos(4w)*s_k - s_{k-4}
    const float n0 = stp * cur0 - prv0;
    const float n1 = stp * cur1 - prv1;
    prv0 = cur0; prv1 = cur1; cur0 = n0; cur1 = n1;
  }

  // All N columns identical; lane L holds M = (L>=16?8:0)+r in acc[r].
  const int r = lane & 7;
  float t0 = (r & 1) ? acc[1] : acc[0];
  float t1 = (r & 1) ? acc[3] : acc[2];
  float t2 = (r & 1) ? acc[5] : acc[4];
  float t3 = (r & 1) ? acc[7] : acc[6];
  float u0 = (r & 2) ? t1 : t0;
  float u1 = (r & 2) ? t3 : t2;
  return (r & 4) ? u1 : u0;
}

// ---------------------------------------------------------------------------
// Kernel 2: one frame per block (512 blocks x 256 threads = 8 waves).
// Each wave covers 256 samples as 8 tile-PAIRS of 32 samples, so that every
// store is full wave-width and the threefry noise uses all 32 lanes.
// ---------------------------------------------------------------------------
__global__ __launch_bounds__(256) void heg_synth_kernel(
    const float* __restrict__ ws, float* __restrict__ out) {
  const int bf = blockIdx.x;              // 0..511
  const int b  = bf >> 7;

  // Stage normalized harmonic amps into LDS.  Prefer the CDNA5 async
  // global->LDS path (ASYNCcnt) when the toolchain exposes it.
  __shared__ float sha[NHARM];
#if __has_builtin(__builtin_amdgcn_global_load_async_to_lds_b32)
  if (threadIdx.x < NHARM) {
    gint_p gp = (gint_p)(ws + WS_HA + bf * NHARM + threadIdx.x);
    lint_p lp = (lint_p)(&sha[threadIdx.x]);
    __builtin_amdgcn_global_load_async_to_lds_b32(gp, lp, 0, 0);
  }
#if __has_builtin(__builtin_amdgcn_s_wait_asynccnt)
  __builtin_amdgcn_s_wait_asynccnt(0);
#endif
  __syncthreads();
#elif __has_builtin(__builtin_amdgcn_load_to_lds)
  if (threadIdx.x < NHARM) {
    gint_p gp = (gint_p)(ws + WS_HA + bf * NHARM + threadIdx.x);
    lint_p lp = (lint_p)(&sha[threadIdx.x]);
    __builtin_amdgcn_load_to_lds(gp, lp, /*size=*/4, /*offset=*/0, /*cpol=*/0);
  }
  __syncthreads();
#else
  if (threadIdx.x < NHARM) sha[threadIdx.x] = ws[WS_HA + bf * NHARM + threadIdx.x];
  __syncthreads();
#endif

  const float dphi = ws[WS_DPHI + bf];
  const float base = ws[WS_BASE + bf];
  const float ap   = ws[WS_AP   + bf];
  const float ld   = ws[WS_LOUD + bf];
  const float lnv  = ws[WS_LN   + bf];

  const int lane = threadIdx.x & 31;
  const int wv   = threadIdx.x >> 5;      // 8 waves, 256 samples each
  const bool hi  = lane >= 16;            // lanes 16..31 hold K=2,3 of A
  const int  m   = lane & 15;

  // Preload B fragments for all 16 K-chunks (broadcast ha across N columns).
  // B 4x16 f32: VGPR0 = rows K=0 (lanes 0-15) / K=2 (lanes 16-31), VGPR1 = K=1/K=3.
  v2f bq[16];
#pragma unroll
  for (int c = 0; c < 16; ++c) {
    bq[c].x = sha[4 * c + (hi ? 2 : 0)];
    bq[c].y = sha[4 * c + (hi ? 3 : 1)];
  }

  const int    frame_t0 = (bf & 127) * BLOCK_SZ;
  const size_t outbase  = (size_t)b * T_PER_B;
  // route sample M=m to lane m: m<8 from lane m, m>=8 from lane m+8
  const int srcLane = (m < 8) ? m : (m + 8);

  for (int p = 0; p < 8; ++p) {
    const int s0 = wv * 256 + p * 32;     // first sample of this pair in frame

    const float x0 = heg_tile_sinsum(base, dphi, s0 + m,      hi, lane, bq);
    const float r0 = __shfl(x0, srcLane, 32);   // lanes 0..15 keep tile-0 samples
    const float x1 = heg_tile_sinsum(base, dphi, s0 + 16 + m, hi, lane, bq);
    const float r1 = __shfl(x1, srcLane, 32);   // lanes 16..31 keep tile-1 samples

    const float sinsum = hi ? r1 : r0;    // lane L -> sample s0 + L

    const int t = frame_t0 + s0 + lane;
    const unsigned g = (unsigned)(b * T_PER_B + t);       // flat [B,T] index
    unsigned bits = heg_threefry_bits(g);
    bits = (bits >> 9) | 0x3f800000u;
    const float u = __uint_as_float(bits) - 1.0f;         // U[0,1)
    const float noise = (u * 2.0f - 1.0f) * GLOBAL_AMP * 0.1f * ap;
    const float sig   = (sinsum * GLOBAL_AMP + noise) * ld;

    __builtin_nontemporal_store(sig,   &out[outbase + t]);
    __builtin_nontemporal_store(noise, &out[(size_t)N_TOTAL + outbase + t]);
    __builtin_nontemporal_store(lnv,   &out[(size_t)2 * N_TOTAL + outbase + t]);
  }
}

// ---------------------------------------------------------------------------
extern "C" void kernel_launch(void* const* d_in, const int* in_sizes, int n_in,
                              void* d_out, int out_size, void* d_ws, size_t ws_size,
                              hipStream_t stream) {
  const float* pitch = (const float*)d_in[0];
  const float* harm  = (const float*)d_in[1];
  const float* per   = (const float*)d_in[2];
  const float* loud  = (const float*)d_in[3];
  const float* lnorm = (const float*)d_in[4];
  float* ws  = (float*)d_ws;
  float* out = (float*)d_out;

  heg_prep_kernel<<<BATCH, FRAMES, 0, stream>>>(pitch, harm, per, loud, lnorm, ws);
  heg_synth_kernel<<<BATCH * FRAMES, 256, 0, stream>>>(ws, out);
}